// GATLayer_27831388078278
// MI455X (gfx1250) — compile-verified
//
#include <hip/hip_runtime.h>
#include <math.h>

typedef __attribute__((ext_vector_type(2))) float v2f;
typedef __attribute__((ext_vector_type(8))) float v8f;

#define IN_DIM   128
#define OUT_DIM  64
#define HEADS    4
#define HW_COLS  (HEADS * OUT_DIM)   // 256
#define SLOPE    0.2f
#define CLIP_LO  0.005f
#define CLIP_HI  10.0f

// ---------------------------------------------------------------------------
// Kernel 1: hw[N,256] = h[N,128] @ Wcat[128,256]  via V_WMMA_F32_16X16X4_F32
// Wcat[k][g*64+j] = W[g][k][j]  (W is [4,128,64] row-major)
// Block = 256 threads = 8 waves; wave (r,c): rows [blk*32+16r, +16), cols [64c, +64)
// ---------------------------------------------------------------------------
__global__ __launch_bounds__(256) void gat_gemm_wmma(
    const float* __restrict__ h, const float* __restrict__ W,
    float* __restrict__ hw, int nrows)
{
    const int wave = threadIdx.x >> 5;
    const int lane = threadIdx.x & 31;
    const int l16  = lane & 15;
    const int hi   = lane >> 4;                 // 0: K=k,k+1  1: K=k+2,k+3
    const int M0   = blockIdx.x * 32 + (wave >> 2) * 16;
    if (M0 >= nrows) return;
    const int colGrp = wave & 3;                // head / 64-col group

    // A fragment source: row = M0 + l16, cols step by 4 with +2*hi offset
    int arow = M0 + l16;
    if (arow >= nrows) arow = nrows - 1;        // defensive clamp (grid is exact for N=100000)
    const float* __restrict__ aptr = h + (size_t)arow * IN_DIM + 2 * hi;

    // B fragment source: Wcat[krow][colGrp*64 + j] = W + colGrp*8192 + krow*64 + j
    const float* __restrict__ bptr = W + (size_t)colGrp * (IN_DIM * OUT_DIM) + l16;

    v8f acc0 = {}; v8f acc1 = {}; v8f acc2 = {}; v8f acc3 = {};

    #pragma unroll 4
    for (int k = 0; k < IN_DIM; k += 4) {
        v2f afrag = *(const v2f*)(aptr + k);
        const float* bk = bptr + (k + 2 * hi) * OUT_DIM;
        v2f b0 = { bk[ 0], bk[ 0 + OUT_DIM] };
        v2f b1 = { bk[16], bk[16 + OUT_DIM] };
        v2f b2 = { bk[32], bk[32 + OUT_DIM] };
        v2f b3 = { bk[48], bk[48 + OUT_DIM] };
        acc0 = __builtin_amdgcn_wmma_f32_16x16x4_f32(false, afrag, false, b0, (short)0, acc0, false, false);
        acc1 = __builtin_amdgcn_wmma_f32_16x16x4_f32(false, afrag, false, b1, (short)0, acc1, false, false);
        acc2 = __builtin_amdgcn_wmma_f32_16x16x4_f32(false, afrag, false, b2, (short)0, acc2, false, false);
        acc3 = __builtin_amdgcn_wmma_f32_16x16x4_f32(false, afrag, false, b3, (short)0, acc3, false, false);
    }

    // C/D layout: VGPR v -> row M0 + v + 8*hi, col = N0 + t*16 + l16
    const int cbase = colGrp * 64 + l16;
    #pragma unroll
    for (int v = 0; v < 8; ++v) {
        int row = M0 + v + 8 * hi;
        if (row < nrows) {
            float* orow = hw + (size_t)row * HW_COLS + cbase;
            orow[ 0] = acc0[v];
            orow[16] = acc1[v];
            orow[32] = acc2[v];
            orow[48] = acc3[v];
        }
    }
}

// ---------------------------------------------------------------------------
// Kernel 2: per-node attention scores  s_src[n,k] = hw[n, k*64: ]·a[k,:64]
//                                      s_dst[n,k] = hw[n, k*64: ]·a[k,64:]
// ---------------------------------------------------------------------------
__global__ __launch_bounds__(256) void gat_scores(
    const float* __restrict__ hw, const float* __restrict__ a,
    float* __restrict__ s_src, float* __restrict__ s_dst, int total)
{
    int t = blockIdx.x * blockDim.x + threadIdx.x;
    if (t >= total) return;
    int n = t >> 2, k = t & 3;
    const float* row = hw + (size_t)n * HW_COLS + k * OUT_DIM;
    const float* a1  = a + k * (2 * OUT_DIM);
    const float* a2  = a1 + OUT_DIM;
    float ss = 0.f, sd = 0.f;
    #pragma unroll
    for (int j = 0; j < OUT_DIM; j += 4) {
        float4 hv = *(const float4*)(row + j);
        float4 v1 = *(const float4*)(a1 + j);
        float4 v2 = *(const float4*)(a2 + j);
        ss += hv.x * v1.x + hv.y * v1.y + hv.z * v1.z + hv.w * v1.w;
        sd += hv.x * v2.x + hv.y * v2.y + hv.z * v2.z + hv.w * v2.w;
    }
    s_src[t] = ss;
    s_dst[t] = sd;
}

__device__ __forceinline__ float edge_weight(float logit)
{
    float lr = logit > 0.f ? logit : SLOPE * logit;
    float ev = __expf(lr);
    return fminf(fmaxf(ev, CLIP_LO), CLIP_HI);
}

// ---------------------------------------------------------------------------
// Kernel 3: softmax denominators  denom[dst,k] += e(edge,k)
// ---------------------------------------------------------------------------
__global__ __launch_bounds__(256) void gat_edge_denom(
    const int* __restrict__ src, const int* __restrict__ dst,
    const float* __restrict__ s_src, const float* __restrict__ s_dst,
    float* __restrict__ denom, int nedges)
{
    int e = blockIdx.x * blockDim.x + threadIdx.x;
    if (e >= nedges) return;
    int s = src[e], d = dst[e];
    #pragma unroll
    for (int k = 0; k < HEADS; ++k) {
        float ev = edge_weight(s_src[s * HEADS + k] + s_dst[d * HEADS + k]);
        atomicAdd(&denom[d * HEADS + k], ev);
    }
}

// ---------------------------------------------------------------------------
// Kernel 4: weighted scatter. Half-wave (16 lanes) per edge; heads summed in
// registers before the atomic so we do 64 (not 256) atomics per edge.
// ---------------------------------------------------------------------------
__global__ __launch_bounds__(256) void gat_edge_scatter(
    const int* __restrict__ src, const int* __restrict__ dst,
    const float* __restrict__ s_src, const float* __restrict__ s_dst,
    const float* __restrict__ denom, const float* __restrict__ hw,
    float* __restrict__ out, int nedges)
{
    int tid = blockIdx.x * blockDim.x + threadIdx.x;
    int e   = tid >> 4;
    int l   = tid & 15;
    if (e >= nedges) return;
    int s = src[e], d = dst[e];

    float alpha[HEADS];
    #pragma unroll
    for (int k = 0; k < HEADS; ++k) {
        float ev = edge_weight(s_src[s * HEADS + k] + s_dst[d * HEADS + k]);
        alpha[k] = ev / denom[d * HEADS + k];
    }

    const float* hrow = hw  + (size_t)s * HW_COLS;
    float*       orow = out + (size_t)d * OUT_DIM;
    #pragma unroll
    for (int c = 0; c < 4; ++c) {
        int dcol = c * 16 + l;
        float m = alpha[0] * hrow[dcol]
                + alpha[1] * hrow[OUT_DIM + dcol]
                + alpha[2] * hrow[2 * OUT_DIM + dcol]
                + alpha[3] * hrow[3 * OUT_DIM + dcol];
        atomicAdd(&orow[dcol], m);
    }
}

// ---------------------------------------------------------------------------
// Kernel 5: mean over heads (divide by 4)
// ---------------------------------------------------------------------------
__global__ __launch_bounds__(256) void gat_scale(float* __restrict__ out, int n)
{
    int i = blockIdx.x * blockDim.x + threadIdx.x;
    if (i < n) out[i] *= 0.25f;
}

// ---------------------------------------------------------------------------
extern "C" void kernel_launch(void* const* d_in, const int* in_sizes, int n_in,
                              void* d_out, int out_size, void* d_ws, size_t ws_size,
                              hipStream_t stream)
{
    const float* h     = (const float*)d_in[0];
    const int*   edges = (const int*)d_in[1];   // harness delivers integer inputs as int32
    const float* W     = (const float*)d_in[2];
    const float* a     = (const float*)d_in[3];

    const int N = in_sizes[0] / IN_DIM;
    const int E = in_sizes[1] / 2;
    const int* src = edges;
    const int* dst = edges + E;
    float* out = (float*)d_out;

    // workspace layout (256B aligned): hw | s_src | s_dst | denom
    char* ws = (char*)d_ws;
    size_t off = 0;
    float* hw = (float*)(ws + off);
    off += ((size_t)N * HW_COLS * sizeof(float) + 255) & ~(size_t)255;
    float* s_src = (float*)(ws + off);
    off += ((size_t)N * HEADS * sizeof(float) + 255) & ~(size_t)255;
    float* s_dst = (float*)(ws + off);
    off += ((size_t)N * HEADS * sizeof(float) + 255) & ~(size_t)255;
    float* denom = (float*)(ws + off);

    // accumulation buffers must start at zero each call
    hipMemsetAsync(out,   0, (size_t)out_size * sizeof(float), stream);
    hipMemsetAsync(denom, 0, (size_t)N * HEADS * sizeof(float), stream);

    gat_gemm_wmma<<<(N + 31) / 32, 256, 0, stream>>>(h, W, hw, N);
    gat_scores<<<(N * HEADS + 255) / 256, 256, 0, stream>>>(hw, a, s_src, s_dst, N * HEADS);
    gat_edge_denom<<<(E + 255) / 256, 256, 0, stream>>>(src, dst, s_src, s_dst, denom, E);
    gat_edge_scatter<<<(E + 15) / 16, 256, 0, stream>>>(src, dst, s_src, s_dst, denom, hw, out, E);
    gat_scale<<<(out_size + 255) / 256, 256, 0, stream>>>(out, out_size);
}